// SoftHd_6476810682382
// MI455X (gfx1250) — compile-verified
//
#include <hip/hip_runtime.h>
#include <hip/hip_bf16.h>
#include <math.h>

#define EMB    2048
#define T_OCR  8192
#define T_Q    2048
#define NB     32
#define N_OCR  256   // ocr rows per batch
#define N_Q    64    // query rows per batch
// LSA runs on the transposed matrix: MROWS x MCOLS = 64 x 256
#define MROWS  64
#define MCOLS  256

typedef __attribute__((ext_vector_type(2))) float v2f;
typedef __attribute__((ext_vector_type(8))) float v8f;

// ---------------------------------------------------------------------------
// Kernel 1: inverse L2 norms of every feature row (ocr rows then query rows).
// One 256-thread block per row, 2048 elements each.
// ---------------------------------------------------------------------------
__global__ void norms_kernel(const float* __restrict__ ocrF,
                             const float* __restrict__ qryF,
                             float* __restrict__ invo,
                             float* __restrict__ invq) {
  const int r   = blockIdx.x;
  const int tid = threadIdx.x;
  const float* src = (r < T_OCR) ? (ocrF + (size_t)r * EMB)
                                 : (qryF + (size_t)(r - T_OCR) * EMB);
  float s = 0.0f;
  for (int k = tid; k < EMB; k += 256) {
    float x = src[k];
    s += x * x;
  }
  __shared__ float red[256];
  red[tid] = s;
  __syncthreads();
  for (int st = 128; st > 0; st >>= 1) {
    if (tid < st) red[tid] += red[tid + st];
    __syncthreads();
  }
  if (tid == 0) {
    float inv = 1.0f / fmaxf(sqrtf(red[0]), 1e-12f);
    if (r < T_OCR) invo[r] = inv;
    else           invq[r - T_OCR] = inv;
  }
}

// ---------------------------------------------------------------------------
// Kernel 2: cost matrices via f32 WMMA.
// One wave (32 threads) per 16x16 output tile.
//   grid = (N_Q/16, N_OCR/16, NB) = (4, 16, 32)
// A (16x4 f32): lanes 0-15 hold M=lane, VGPR0/1 = K0/K1; lanes 16-31 = K2/K3.
// B (4x16 f32): lanes 0-15 hold N=lane, VGPR0/1 = K0/K1; lanes 16-31 = K2/K3.
// D (16x16 f32): VGPR r, lanes 0-15 -> M=r, lanes 16-31 -> M=r+8; N=lane&15.
// Normalization folded in as post-scale by inverse norms; result written
// TRANSPOSED (query-major, 64x256) for the LSA solver.
// ---------------------------------------------------------------------------
__global__ void cost_wmma_kernel(const float* __restrict__ ocrF,
                                 const float* __restrict__ qryF,
                                 const int*  __restrict__ oidx,
                                 const int*  __restrict__ qidx,
                                 const float* __restrict__ invo,
                                 const float* __restrict__ invq,
                                 float* __restrict__ costT) {
  const int b    = blockIdx.z;
  const int m0   = blockIdx.y * 16;   // ocr-row tile base (0..255)
  const int n0   = blockIdx.x * 16;   // query-row tile base (0..63)
  const int lane = threadIdx.x;       // 0..31, full wave, uniform EXEC
  const int half = lane >> 4;         // 0 or 1
  const int ll   = lane & 15;
  const int kh   = half * 2;          // K sub-offset for this half-wave

  const int arow = oidx[b * N_OCR + m0 + ll];
  const int brow = qidx[b * N_Q   + n0 + ll];
  const float* aptr = ocrF + (size_t)arow * EMB + kh;
  const float* bptr = qryF + (size_t)brow * EMB + kh;

  v8f acc = {};
#pragma unroll 8
  for (int k = 0; k < EMB; k += 4) {
    v2f av = *(const v2f*)(aptr + k);
    v2f bv = *(const v2f*)(bptr + k);
    // D = A(16x4) * B(4x16) + C, f32 in / f32 accumulate
    acc = __builtin_amdgcn_wmma_f32_16x16x4_f32(
        /*neg_a=*/false, av, /*neg_b=*/false, bv,
        /*c_mod=*/(short)0, acc, /*reuse_a=*/false, /*reuse_b=*/false);
  }

  const int   N  = n0 + ll;                       // query index in batch
  const float sq = invq[qidx[b * N_Q + N]];
#pragma unroll
  for (int r = 0; r < 8; ++r) {
    const int   M  = m0 + r + half * 8;           // ocr index in batch
    const float so = invo[oidx[b * N_OCR + M]];
    const float simv = acc[r] * so * sq;
    // transposed layout: costT[b][N][M]
    costT[((size_t)b * MROWS + N) * MCOLS + M] = 1.0f - fabsf(simv);
  }
}

// ---------------------------------------------------------------------------
// Kernel 3: Jonker-Volgenant LSA (float64, matching the reference solver),
// one workgroup (256 threads) per batch on the 64x256 transposed cost.
// Inner column sweep, first-occurrence argmin, and dual updates are parallel;
// the augmenting-path rewind is serial on thread 0.
// ---------------------------------------------------------------------------
__global__ void lsa_kernel(const float* __restrict__ costAll,
                           float* __restrict__ out) {
  const int b   = blockIdx.x;
  const int tid = threadIdx.x;                    // 0..255, one per column
  const float* C = costAll + (size_t)b * MROWS * MCOLS;
  const double INFD = 1e300;

  __shared__ double u[MROWS];
  __shared__ double v[MCOLS];
  __shared__ double minv[MCOLS];
  __shared__ int    way[MCOLS];
  __shared__ int    p[MCOLS + 1];
  __shared__ int    used[MCOLS + 1];
  __shared__ double redv[MCOLS];
  __shared__ int    redi[MCOLS];
  __shared__ int    s_j0;

  if (tid < MROWS) u[tid] = 0.0;
  v[tid] = 0.0;
  p[tid] = -1;
  if (tid == 0) p[MCOLS] = -1;
  __syncthreads();

  for (int i = 0; i < MROWS; ++i) {
    if (tid == 0) { p[MCOLS] = i; s_j0 = MCOLS; }
    minv[tid] = INFD;
    way[tid]  = MCOLS;
    used[tid] = 0;
    if (tid == 0) used[MCOLS] = 0;
    __syncthreads();

    while (true) {
      if (tid == 0) used[s_j0] = 1;
      __syncthreads();
      const int    j0  = s_j0;
      const int    i0  = p[j0];
      const double ui0 = u[i0];

      if (!used[tid]) {
        double cur = (double)C[i0 * MCOLS + tid] - ui0 - v[tid];
        if (cur < minv[tid]) { minv[tid] = cur; way[tid] = j0; }
        redv[tid] = minv[tid];
      } else {
        redv[tid] = INFD;            // masked (used) columns
      }
      redi[tid] = tid;
      __syncthreads();

      // lexicographic (value, index) min == np.argmin first occurrence
      for (int s = 128; s > 0; s >>= 1) {
        if (tid < s) {
          double a = redv[tid], c = redv[tid + s];
          int    ia = redi[tid], ic = redi[tid + s];
          if (c < a || (c == a && ic < ia)) { redv[tid] = c; redi[tid] = ic; }
        }
        __syncthreads();
      }
      const double delta = redv[0];
      const int    j1    = redi[0];

      // dual updates (used columns have p[] >= 0 and distinct rows)
      if (used[tid]) {
        u[p[tid]] += delta;
        v[tid]    -= delta;
      } else {
        minv[tid] -= delta;
      }
      if (tid == 0) { u[i] += delta; s_j0 = j1; }
      __syncthreads();
      if (p[j1] == -1) break;
    }

    // augmenting-path rewind (serial)
    if (tid == 0) {
      int j0 = s_j0;
      while (j0 != MCOLS) {
        int j1 = way[j0];
        p[j0] = p[j1];
        j0 = j1;
      }
    }
    __syncthreads();
  }

  // mean of assigned costs: sum_j (p[j] >= 0) C[p[j]][j] / 64
  redv[tid] = (p[tid] >= 0) ? (double)C[p[tid] * MCOLS + tid] : 0.0;
  __syncthreads();
  for (int s = 128; s > 0; s >>= 1) {
    if (tid < s) redv[tid] += redv[tid + s];
    __syncthreads();
  }
  if (tid == 0) out[b] = (float)(redv[0] / (double)MROWS);
}

// ---------------------------------------------------------------------------
extern "C" void kernel_launch(void* const* d_in, const int* in_sizes, int n_in,
                              void* d_out, int out_size, void* d_ws, size_t ws_size,
                              hipStream_t stream) {
  const float* ocrF = (const float*)d_in[0];   // (8192, 2048) f32
  const float* qryF = (const float*)d_in[1];   // (2048, 2048) f32
  const int*   oidx = (const int*)d_in[2];     // (32, 256) i32
  const int*   qidx = (const int*)d_in[3];     // (32, 64)  i32
  float*       out  = (float*)d_out;           // (32,) f32

  float* invo  = (float*)d_ws;                 // 8192 floats
  float* invq  = invo + T_OCR;                 // 2048 floats
  float* costT = invq + T_Q;                   // 32*64*256 floats (transposed cost)

  norms_kernel<<<T_OCR + T_Q, 256, 0, stream>>>(ocrF, qryF, invo, invq);

  dim3 gB(N_Q / 16, N_OCR / 16, NB);           // (4, 16, 32)
  cost_wmma_kernel<<<gB, 32, 0, stream>>>(ocrF, qryF, oidx, qidx, invo, invq, costT);

  lsa_kernel<<<NB, 256, 0, stream>>>(costT, out);
}